// NequIPMessagePassingLayer_50002009260583
// MI455X (gfx1250) — compile-verified
//
#include <hip/hip_runtime.h>
#include <hip/hip_bf16.h>

typedef float v2f __attribute__((ext_vector_type(2)));
typedef float v8f __attribute__((ext_vector_type(8)));

__device__ __forceinline__ float silu_f(float x) {
    return x / (1.0f + __expf(-x));
}

__device__ __forceinline__ void atomic_add_f32(float* p, float v) {
    __hip_atomic_fetch_add(p, v, __ATOMIC_RELAXED, __HIP_MEMORY_SCOPE_AGENT);
}

// ---------------------------------------------------------------------------
// Zero the accumulator region of the workspace (a0 | a1).
// ---------------------------------------------------------------------------
__global__ void zero_kernel(float* __restrict__ p, size_t n) {
    size_t i = (size_t)blockIdx.x * blockDim.x + threadIdx.x;
    size_t stride = (size_t)gridDim.x * blockDim.x;
    for (; i < n; i += stride) p[i] = 0.0f;
}

// ---------------------------------------------------------------------------
// Node pre-transform:  p0 = x0 @ W_pre0   (N x 32)
//                      p1[:, :, d] = x1[:, :, d] @ W_pre1  (N x 32 x 3)
// One wave (32 lanes) handles 16 nodes using V_WMMA_F32_16X16X4_F32.
// ---------------------------------------------------------------------------
__global__ __launch_bounds__(256) void node_pre_kernel(
    const float* __restrict__ emb,    // (N, 128)
    const float* __restrict__ Wpre0,  // (32, 32) row-major
    const float* __restrict__ Wpre1,  // (32, 32) row-major
    float* __restrict__ p0,           // (N, 32)
    float* __restrict__ p1,           // (N, 32, 3)
    int N)
{
    const int lane = threadIdx.x & 31;
    const int wv   = threadIdx.x >> 5;
    const int base = (blockIdx.x * 8 + wv) * 16;
    if (base >= N) return;                    // wave-uniform branch

    const bool fullTile = (base + 16 <= N);   // wave-uniform

    const int row = lane & 15;                // A-matrix row / C-matrix col
    const int khi = lane >> 4;                // lane-half selector
    const int col = row;

    int nodeA = base + row;
    if (nodeA > N - 1) nodeA = N - 1;         // clamp (keeps EXEC full)
    const float* arow = emb + (size_t)nodeA * 128;

    // ---- p0 = x0 @ W_pre0 ----
    v8f acc0 = {}, acc1 = {};
#pragma unroll
    for (int kb = 0; kb < 32; kb += 4) {
        const int k0 = kb + khi * 2;
        v2f a;  a.x = arow[k0];              a.y = arow[k0 + 1];
        v2f b0; b0.x = Wpre0[k0 * 32 + col]; b0.y = Wpre0[(k0 + 1) * 32 + col];
        v2f b1; b1.x = Wpre0[k0 * 32 + 16 + col];
                b1.y = Wpre0[(k0 + 1) * 32 + 16 + col];
        acc0 = __builtin_amdgcn_wmma_f32_16x16x4_f32(false, a, false, b0, (short)0, acc0, false, false);
        acc1 = __builtin_amdgcn_wmma_f32_16x16x4_f32(false, a, false, b1, (short)0, acc1, false, false);
    }
    if (fullTile) {
#pragma unroll
        for (int v = 0; v < 8; ++v) {
            const int node = base + v + 8 * khi;
            p0[(size_t)node * 32 + col]      = acc0[v];
            p0[(size_t)node * 32 + 16 + col] = acc1[v];
        }
    } else {
#pragma unroll
        for (int v = 0; v < 8; ++v) {
            const int node = base + v + 8 * khi;
            if (node < N) {
                p0[(size_t)node * 32 + col]      = acc0[v];
                p0[(size_t)node * 32 + 16 + col] = acc1[v];
            }
        }
    }

    // ---- p1[:, :, d] = x1[:, :, d] @ W_pre1 ----
#pragma unroll
    for (int d = 0; d < 3; ++d) {
        v8f c0 = {}, c1 = {};
#pragma unroll
        for (int kb = 0; kb < 32; kb += 4) {
            const int k0 = kb + khi * 2;
            v2f a;  a.x = arow[32 + k0 * 3 + d];       a.y = arow[32 + (k0 + 1) * 3 + d];
            v2f b0; b0.x = Wpre1[k0 * 32 + col];       b0.y = Wpre1[(k0 + 1) * 32 + col];
            v2f b1; b1.x = Wpre1[k0 * 32 + 16 + col];  b1.y = Wpre1[(k0 + 1) * 32 + 16 + col];
            c0 = __builtin_amdgcn_wmma_f32_16x16x4_f32(false, a, false, b0, (short)0, c0, false, false);
            c1 = __builtin_amdgcn_wmma_f32_16x16x4_f32(false, a, false, b1, (short)0, c1, false, false);
        }
        if (fullTile) {
#pragma unroll
            for (int v = 0; v < 8; ++v) {
                const int node = base + v + 8 * khi;
                p1[(size_t)node * 96 + col * 3 + d]        = c0[v];
                p1[(size_t)node * 96 + (16 + col) * 3 + d] = c1[v];
            }
        } else {
#pragma unroll
            for (int v = 0; v < 8; ++v) {
                const int node = base + v + 8 * khi;
                if (node < N) {
                    p1[(size_t)node * 96 + col * 3 + d]        = c0[v];
                    p1[(size_t)node * 96 + (16 + col) * 3 + d] = c1[v];
                }
            }
        }
    }
}

// ---------------------------------------------------------------------------
// Edge kernel: radial MLP (weights in LDS) + envelope + message formation +
// fp32 atomic scatter into a0 (N,64) and a1 (N,64,3). One thread per edge.
// ---------------------------------------------------------------------------
__global__ __launch_bounds__(256) void edge_kernel(
    const float* __restrict__ dist,   // (E,)
    const float* __restrict__ eemb,   // (E, 4)
    const int*   __restrict__ eidx,   // (2, E)
    const float* __restrict__ W1, const float* __restrict__ b1,
    const float* __restrict__ W2, const float* __restrict__ b2,
    const float* __restrict__ W3, const float* __restrict__ b3,
    const float* __restrict__ p0,     // (N, 32)
    const float* __restrict__ p1,     // (N, 32, 3)
    float* __restrict__ a0,           // (N, 64)
    float* __restrict__ a1,           // (N, 64, 3)
    int E)
{
    __shared__ float sW1[64], sB1[8], sW2[64], sB2[8], sW3[1024], sB3[128];
    for (int i = threadIdx.x; i < 64; i += 256) { sW1[i] = W1[i]; sW2[i] = W2[i]; }
    if (threadIdx.x < 8) { sB1[threadIdx.x] = b1[threadIdx.x]; sB2[threadIdx.x] = b2[threadIdx.x]; }
    for (int i = threadIdx.x; i < 1024; i += 256) sW3[i] = W3[i];
    for (int i = threadIdx.x; i < 128;  i += 256) sB3[i] = b3[i];
    __syncthreads();

    const int e = blockIdx.x * 256 + threadIdx.x;
    if (e >= E) return;

    const int ctr = eidx[e];
    const int nbr = eidx[E + e];
    const float r = dist[e];
    const float y0  = eemb[e * 4 + 0];
    const float y1x = eemb[e * 4 + 1];
    const float y1y = eemb[e * 4 + 2];
    const float y1z = eemb[e * 4 + 3];

    // Bessel features
    const float CUT = 5.0f;
    const float invr = 1.0f / r;
    const float wphase = 3.14159265358979f * r / CUT;
    const float c0 = sqrtf(2.0f / CUT);
    float bes[8];
#pragma unroll
    for (int i = 0; i < 8; ++i) bes[i] = c0 * __sinf((float)(i + 1) * wphase) * invr;

    // 8->8->8 MLP (SiLU)
    float h1v[8];
#pragma unroll
    for (int j = 0; j < 8; ++j) {
        float acc = sB1[j];
#pragma unroll
        for (int i = 0; i < 8; ++i) acc += bes[i] * sW1[i * 8 + j];
        h1v[j] = silu_f(acc);
    }
    float h2v[8];
#pragma unroll
    for (int j = 0; j < 8; ++j) {
        float acc = sB2[j];
#pragma unroll
        for (int i = 0; i < 8; ++i) acc += h1v[i] * sW2[i * 8 + j];
        h2v[j] = silu_f(acc);
    }

    // polynomial envelope, p = 6
    const float xr = r / CUT;
    const float xr3 = xr * xr * xr;
    const float xr6 = xr3 * xr3;
    const float xr7 = xr6 * xr;
    const float xr8 = xr7 * xr;
    const float env = (xr < 1.0f) ? (1.0f - 28.0f * xr6 + 48.0f * xr7 - 21.0f * xr8) : 0.0f;

    const float inv_sqrt3 = 0.57735026918962576f;
    const float* x0e = p0 + (size_t)nbr * 32;
    const float* x1e = p1 + (size_t)nbr * 96;
    float* a0c = a0 + (size_t)ctr * 64;
    float* a1c = a1 + (size_t)ctr * 192;

    for (int m = 0; m < 32; ++m) {
        float w00 = sB3[m], w11 = sB3[m + 32], w01 = sB3[m + 64], w10 = sB3[m + 96];
#pragma unroll
        for (int i = 0; i < 8; ++i) {
            const float hv = h2v[i];
            const float* wr = sW3 + i * 128 + m;
            w00 += hv * wr[0];
            w11 += hv * wr[32];
            w01 += hv * wr[64];
            w10 += hv * wr[96];
        }
        w00 *= env; w11 *= env; w01 *= env; w10 *= env;

        const float x0v = x0e[m];
        const float xd0 = x1e[m * 3 + 0];
        const float xd1 = x1e[m * 3 + 1];
        const float xd2 = x1e[m * 3 + 2];

        const float m0a = w00 * x0v * y0;
        const float m0b = w11 * (xd0 * y1x + xd1 * y1y + xd2 * y1z) * inv_sqrt3;
        atomic_add_f32(&a0c[m],      m0a);
        atomic_add_f32(&a0c[32 + m], m0b);

        const float s01 = w01 * x0v;
        const float s10 = w10 * y0;
        atomic_add_f32(&a1c[m * 3 + 0], s01 * y1x);
        atomic_add_f32(&a1c[m * 3 + 1], s01 * y1y);
        atomic_add_f32(&a1c[m * 3 + 2], s01 * y1z);
        atomic_add_f32(&a1c[96 + m * 3 + 0], s10 * xd0);
        atomic_add_f32(&a1c[96 + m * 3 + 1], s10 * xd1);
        atomic_add_f32(&a1c[96 + m * 3 + 2], s10 * xd2);
    }
}

// ---------------------------------------------------------------------------
// Node post-transform + gating. One wave handles 16 nodes with WMMA f32.
//   h0 = a0 @ W_post0 + x0 @ W_si0          (N x 64)
//   h1_d = a1_d @ W_post1 + x1_d @ W_si1    (N x 32 x 3)
//   out = [silu(h0[:, :32]), (silu(h0[:, 32:])[:, :, None] * h1).reshape(N,96)]
// ---------------------------------------------------------------------------
__global__ __launch_bounds__(256) void node_post_kernel(
    const float* __restrict__ emb,     // (N, 128)
    const float* __restrict__ a0,      // (N, 64)
    const float* __restrict__ a1,      // (N, 64, 3)
    const float* __restrict__ Wpost0,  // (64, 64)
    const float* __restrict__ Wpost1,  // (64, 32)
    const float* __restrict__ Wsi0,    // (32, 64)
    const float* __restrict__ Wsi1,    // (32, 32)
    float* __restrict__ out,           // (N, 128)
    int N)
{
    const int lane = threadIdx.x & 31;
    const int wv   = threadIdx.x >> 5;
    const int base = (blockIdx.x * 8 + wv) * 16;
    if (base >= N) return;

    const bool fullTile = (base + 16 <= N);   // wave-uniform

    const int row = lane & 15;
    const int khi = lane >> 4;
    const int col = row;

    int nodeA = base + row;
    if (nodeA > N - 1) nodeA = N - 1;
    const float* a0row = a0 + (size_t)nodeA * 64;
    const float* a1row = a1 + (size_t)nodeA * 192;
    const float* xrow  = emb + (size_t)nodeA * 128;

    // ---- h0 (4 column tiles of 16) ----
    v8f h0acc[4] = {v8f{}, v8f{}, v8f{}, v8f{}};
#pragma unroll
    for (int kb = 0; kb < 64; kb += 4) {            // a0 @ W_post0
        const int k0 = kb + khi * 2;
        v2f a; a.x = a0row[k0]; a.y = a0row[k0 + 1];
#pragma unroll
        for (int t = 0; t < 4; ++t) {
            v2f b; b.x = Wpost0[k0 * 64 + t * 16 + col];
                   b.y = Wpost0[(k0 + 1) * 64 + t * 16 + col];
            h0acc[t] = __builtin_amdgcn_wmma_f32_16x16x4_f32(false, a, false, b, (short)0, h0acc[t], false, false);
        }
    }
#pragma unroll
    for (int kb = 0; kb < 32; kb += 4) {            // x0 @ W_si0
        const int k0 = kb + khi * 2;
        v2f a; a.x = xrow[k0]; a.y = xrow[k0 + 1];
#pragma unroll
        for (int t = 0; t < 4; ++t) {
            v2f b; b.x = Wsi0[k0 * 64 + t * 16 + col];
                   b.y = Wsi0[(k0 + 1) * 64 + t * 16 + col];
            h0acc[t] = __builtin_amdgcn_wmma_f32_16x16x4_f32(false, a, false, b, (short)0, h0acc[t], false, false);
        }
    }

    // scalars out + gate cache (all lane-local: same (row, col) mapping)
    float g0v[8], g1v[8];
#pragma unroll
    for (int v = 0; v < 8; ++v) {
        g0v[v] = silu_f(h0acc[2][v]);
        g1v[v] = silu_f(h0acc[3][v]);
    }
    if (fullTile) {
#pragma unroll
        for (int v = 0; v < 8; ++v) {
            const int node = base + v + 8 * khi;
            float* orow = out + (size_t)node * 128;
            __builtin_nontemporal_store(silu_f(h0acc[0][v]), &orow[col]);
            __builtin_nontemporal_store(silu_f(h0acc[1][v]), &orow[16 + col]);
        }
    } else {
#pragma unroll
        for (int v = 0; v < 8; ++v) {
            const int node = base + v + 8 * khi;
            if (node < N) {
                float* orow = out + (size_t)node * 128;
                orow[col]      = silu_f(h0acc[0][v]);
                orow[16 + col] = silu_f(h0acc[1][v]);
            }
        }
    }

    // ---- h1 per Cartesian component, gated write ----
#pragma unroll
    for (int d = 0; d < 3; ++d) {
        v8f c0 = {}, c1 = {};
#pragma unroll
        for (int kb = 0; kb < 64; kb += 4) {        // a1_d @ W_post1
            const int k0 = kb + khi * 2;
            v2f a; a.x = a1row[k0 * 3 + d]; a.y = a1row[(k0 + 1) * 3 + d];
            v2f b0; b0.x = Wpost1[k0 * 32 + col];      b0.y = Wpost1[(k0 + 1) * 32 + col];
            v2f b1; b1.x = Wpost1[k0 * 32 + 16 + col]; b1.y = Wpost1[(k0 + 1) * 32 + 16 + col];
            c0 = __builtin_amdgcn_wmma_f32_16x16x4_f32(false, a, false, b0, (short)0, c0, false, false);
            c1 = __builtin_amdgcn_wmma_f32_16x16x4_f32(false, a, false, b1, (short)0, c1, false, false);
        }
#pragma unroll
        for (int kb = 0; kb < 32; kb += 4) {        // x1_d @ W_si1
            const int k0 = kb + khi * 2;
            v2f a; a.x = xrow[32 + k0 * 3 + d]; a.y = xrow[32 + (k0 + 1) * 3 + d];
            v2f b0; b0.x = Wsi1[k0 * 32 + col];      b0.y = Wsi1[(k0 + 1) * 32 + col];
            v2f b1; b1.x = Wsi1[k0 * 32 + 16 + col]; b1.y = Wsi1[(k0 + 1) * 32 + 16 + col];
            c0 = __builtin_amdgcn_wmma_f32_16x16x4_f32(false, a, false, b0, (short)0, c0, false, false);
            c1 = __builtin_amdgcn_wmma_f32_16x16x4_f32(false, a, false, b1, (short)0, c1, false, false);
        }
        if (fullTile) {
#pragma unroll
            for (int v = 0; v < 8; ++v) {
                const int node = base + v + 8 * khi;
                float* orow = out + (size_t)node * 128;
                __builtin_nontemporal_store(g0v[v] * c0[v], &orow[32 + col * 3 + d]);
                __builtin_nontemporal_store(g1v[v] * c1[v], &orow[32 + (16 + col) * 3 + d]);
            }
        } else {
#pragma unroll
            for (int v = 0; v < 8; ++v) {
                const int node = base + v + 8 * khi;
                if (node < N) {
                    float* orow = out + (size_t)node * 128;
                    orow[32 + col * 3 + d]        = g0v[v] * c0[v];
                    orow[32 + (16 + col) * 3 + d] = g1v[v] * c1[v];
                }
            }
        }
    }
}

// ---------------------------------------------------------------------------
extern "C" void kernel_launch(void* const* d_in, const int* in_sizes, int n_in,
                              void* d_out, int out_size, void* d_ws, size_t ws_size,
                              hipStream_t stream) {
    const float* emb    = (const float*)d_in[0];   // node_embeddings (N,128)
    // d_in[1] = Z_embeddings (unused by reference)
    const float* dist   = (const float*)d_in[2];   // (E,)
    const float* eemb   = (const float*)d_in[3];   // (E,4)
    const int*   eidx   = (const int*)  d_in[4];   // (2,E)
    const float* Wpre0  = (const float*)d_in[5];
    const float* Wpre1  = (const float*)d_in[6];
    const float* W1     = (const float*)d_in[7];
    const float* b1     = (const float*)d_in[8];
    const float* W2     = (const float*)d_in[9];
    const float* b2     = (const float*)d_in[10];
    const float* W3     = (const float*)d_in[11];
    const float* b3     = (const float*)d_in[12];
    const float* Wpost0 = (const float*)d_in[13];
    const float* Wpost1 = (const float*)d_in[14];
    const float* Wsi0   = (const float*)d_in[15];
    const float* Wsi1   = (const float*)d_in[16];

    const int N = in_sizes[0] / 128;
    const int E = in_sizes[2];

    float* ws = (float*)d_ws;
    float* p0 = ws;                        // N*32
    float* p1 = ws + (size_t)N * 32;       // N*96
    float* a0 = ws + (size_t)N * 128;      // N*64
    float* a1 = ws + (size_t)N * 192;      // N*192

    // zero accumulators (a0|a1 contiguous: N*256 floats)
    zero_kernel<<<2048, 256, 0, stream>>>(ws + (size_t)N * 128, (size_t)N * 256);

    const int nodeBlocks = (N + 127) / 128;
    node_pre_kernel<<<nodeBlocks, 256, 0, stream>>>(emb, Wpre0, Wpre1, p0, p1, N);

    edge_kernel<<<(E + 255) / 256, 256, 0, stream>>>(dist, eemb, eidx,
                                                     W1, b1, W2, b2, W3, b3,
                                                     p0, p1, a0, a1, E);

    node_post_kernel<<<nodeBlocks, 256, 0, stream>>>(emb, a0, a1,
                                                     Wpost0, Wpost1, Wsi0, Wsi1,
                                                     (float*)d_out, N);
}